// MessagePassing_39960375722434
// MI455X (gfx1250) — compile-verified
//
#include <hip/hip_runtime.h>
#include <math.h>

#define IN_FEATS 128
#define HEADS 8
#define HIDDEN 16
#define DIM 128          // HEADS*HIDDEN == IN_FEATS == 128
#define SLOPE 0.2f

typedef __attribute__((ext_vector_type(2))) float v2f;
typedef __attribute__((ext_vector_type(8))) float v8f;

// ---- monotone float<->uint encoding so atomicMax(u32) orders floats ----
__device__ __forceinline__ unsigned fenc(float x) {
    unsigned u = __float_as_uint(x);
    return (u & 0x80000000u) ? ~u : (u | 0x80000000u);
}
__device__ __forceinline__ float fdec(unsigned u) {
    return __uint_as_float((u & 0x80000000u) ? (u & 0x7fffffffu) : ~u);
}

__device__ __forceinline__ float leaky(float x) { return x > 0.f ? x : SLOPE * x; }

// Wave-uniform-guarded tile store: fast path (full tile) has no EXEC juggling.
__device__ __forceinline__ void store_tile(float* __restrict__ H, v8f c, int tr,
                                           int col, int lane, int n) {
    if (tr * 16 >= n) return;                       // wave-uniform
    int mbase = tr * 16 + ((lane >> 4) << 3);       // lanes 16-31 -> M=j+8
    if (tr * 16 + 16 <= n) {                        // wave-uniform full tile
#pragma unroll
        for (int j = 0; j < 8; ++j)
            H[(long)(mbase + j) * DIM + col] = c[j];
    } else {                                        // boundary tile (rare)
#pragma unroll
        for (int j = 0; j < 8; ++j)
            if (mbase + j < n) H[(long)(mbase + j) * DIM + col] = c[j];
    }
}

// ============================================================================
// Kernel 1: H = X @ W  ([n,128] @ [128,128]) via V_WMMA_F32_16X16X4_F32.
// One wave computes a 64x16 output block (4 M-tiles), reusing each B fragment
// across 4 WMMAs. K-loop: 32 steps x 4 WMMA = 128 WMMA per wave.
// ============================================================================
__global__ __launch_bounds__(128) void gat_gemm_wmma(
    const float* __restrict__ X, const float* __restrict__ W,
    float* __restrict__ H, int n) {
    int wave = (blockIdx.x * blockDim.x + threadIdx.x) >> 5;
    int lane = threadIdx.x & 31;
    int nRowTiles  = (n + 15) >> 4;
    int nRowGroups = (nRowTiles + 3) >> 2;
    int group   = wave >> 3;      // 64-row group
    int tileCol = wave & 7;       // 8 col tiles of 16
    if (group >= nRowGroups) return;

    const int row  = lane & 15;           // M (A) / N (B,C,D)
    const int kOff = (lane >> 4) << 1;    // lanes 16-31 carry K+2,K+3
    const int baseTile = group * 4;

    // clamp A row pointers so EXEC stays uniform; stores are guarded instead
    const float* xr[4];
#pragma unroll
    for (int i = 0; i < 4; ++i) {
        int arow = (baseTile + i) * 16 + row;
        if (arow >= n) arow = n - 1;
        xr[i] = X + (long)arow * DIM;
    }
    const float* wcol = W + tileCol * 16 + row;

    v8f c0 = {}, c1 = {}, c2 = {}, c3 = {};
#pragma unroll 4
    for (int k = 0; k < DIM; k += 4) {
        v2f b;
        b.x = wcol[(k + kOff) * DIM];
        b.y = wcol[(k + kOff + 1) * DIM];
        v2f a0, a1, a2, a3;
        a0.x = xr[0][k + kOff]; a0.y = xr[0][k + kOff + 1];
        a1.x = xr[1][k + kOff]; a1.y = xr[1][k + kOff + 1];
        a2.x = xr[2][k + kOff]; a2.y = xr[2][k + kOff + 1];
        a3.x = xr[3][k + kOff]; a3.y = xr[3][k + kOff + 1];
        // (neg_a, A, neg_b, B, c_mod, C, reuse_a, reuse_b)
        c0 = __builtin_amdgcn_wmma_f32_16x16x4_f32(false, a0, false, b, (short)0, c0, false, false);
        c1 = __builtin_amdgcn_wmma_f32_16x16x4_f32(false, a1, false, b, (short)0, c1, false, false);
        c2 = __builtin_amdgcn_wmma_f32_16x16x4_f32(false, a2, false, b, (short)0, c2, false, false);
        c3 = __builtin_amdgcn_wmma_f32_16x16x4_f32(false, a3, false, b, (short)0, c3, false, false);
    }

    int col = tileCol * 16 + row;
    store_tile(H, c0, baseTile + 0, col, lane, n);
    store_tile(H, c1, baseTile + 1, col, lane, n);
    store_tile(H, c2, baseTile + 2, col, lane, n);
    store_tile(H, c3, baseTile + 3, col, lane, n);
}

// ============================================================================
// Kernel 2: per (node,head): el/er dots, init m=-inf(enc), denom=0, zero out.
// ============================================================================
__global__ void gat_node_prep(const float* __restrict__ H,
                              const float* __restrict__ al,
                              const float* __restrict__ ar,
                              float* __restrict__ el, float* __restrict__ er,
                              unsigned* __restrict__ m, float* __restrict__ denom,
                              float* __restrict__ outZero, int n) {
    int t = blockIdx.x * blockDim.x + threadIdx.x;
    if (t >= n * HEADS) return;
    int node = t >> 3, h = t & 7;
    const float* hp  = H + (long)node * DIM + h * HIDDEN;
    const float* alp = al + h * HIDDEN;
    const float* arp = ar + h * HIDDEN;
    float sl = 0.f, sr = 0.f;
#pragma unroll
    for (int f = 0; f < HIDDEN; ++f) { float v = hp[f]; sl += v * alp[f]; sr += v * arp[f]; }
    el[t] = sl;
    er[t] = sr;
    m[t] = fenc(-3.0e38f);
    denom[t] = 0.f;
    float* op = outZero + (long)node * DIM + h * HIDDEN;
#pragma unroll
    for (int f = 0; f < HIDDEN; ++f) op[f] = 0.f;
}

// ============================================================================
// Kernel 3: segment-max of leaky(el[src]+er[dst]) into m[dst] (encoded u32).
// ============================================================================
__global__ void gat_edge_max(const int* __restrict__ src, const int* __restrict__ dst,
                             const float* __restrict__ el, const float* __restrict__ er,
                             unsigned* __restrict__ m, int nE) {
    int t = blockIdx.x * blockDim.x + threadIdx.x;
    if (t >= nE * HEADS) return;
    int e = t >> 3, h = t & 7;
    int s = src[e], d = dst[e];
    float ee = leaky(el[s * HEADS + h] + er[d * HEADS + h]);
    atomicMax(&m[d * HEADS + h], fenc(ee));
}

// ============================================================================
// Kernel 4: denom[dst] += exp(e - m[dst])
// ============================================================================
__global__ void gat_edge_sum(const int* __restrict__ src, const int* __restrict__ dst,
                             const float* __restrict__ el, const float* __restrict__ er,
                             const unsigned* __restrict__ m, float* __restrict__ denom,
                             int nE) {
    int t = blockIdx.x * blockDim.x + threadIdx.x;
    if (t >= nE * HEADS) return;
    int e = t >> 3, h = t & 7;
    int s = src[e], d = dst[e];
    float ee = leaky(el[s * HEADS + h] + er[d * HEADS + h]);
    float ex = expf(ee - fdec(m[d * HEADS + h]));
    atomicAdd(&denom[d * HEADS + h], ex);
}

// ============================================================================
// Kernel 5: one wave32 per edge. Lanes 0-7 compute alpha per head, broadcast
// via shuffle; each lane scatters 4 coalesced floats of h[src]*alpha to dst.
// ============================================================================
__global__ __launch_bounds__(256) void gat_message(
    const float* __restrict__ H, const int* __restrict__ src,
    const int* __restrict__ dst, const float* __restrict__ el,
    const float* __restrict__ er, const unsigned* __restrict__ m,
    const float* __restrict__ denom, float* __restrict__ out, int nE) {
    int wave = (blockIdx.x * blockDim.x + threadIdx.x) >> 5;
    int lane = threadIdx.x & 31;
    if (wave >= nE) return;
    int s = src[wave], d = dst[wave];
    float a = 0.f;
    if (lane < HEADS) {
        float ee = leaky(el[s * HEADS + lane] + er[d * HEADS + lane]);
        float ex = expf(ee - fdec(m[d * HEADS + lane]));
        float dn = denom[d * HEADS + lane];
        a = ex / (dn > 0.f ? dn : 1.f);
    }
    const float* hp = H + (long)s * DIM;
    float* op = out + (long)d * DIM;
#pragma unroll
    for (int k = 0; k < 4; ++k) {
        int f = lane + 32 * k;
        float av = __shfl(a, f >> 4, 32);   // head = f/16
        atomicAdd(&op[f], hp[f] * av);
    }
}

// ============================================================================
static void run_layer(const float* X, const float* W, const float* al,
                      const float* ar, const int* src, const int* dst,
                      float* hbuf, float* el, float* er, unsigned* m,
                      float* denom, float* out, int N, int E,
                      hipStream_t stream) {
    int nRowTiles  = (N + 15) / 16;
    int nRowGroups = (nRowTiles + 3) / 4;
    int waves      = nRowGroups * 8;
    int gemmBlocks = (waves * 32 + 127) / 128;
    gat_gemm_wmma<<<gemmBlocks, 128, 0, stream>>>(X, W, hbuf, N);
    gat_node_prep<<<(N * HEADS + 255) / 256, 256, 0, stream>>>(
        hbuf, al, ar, el, er, m, denom, out, N);
    gat_edge_max<<<(E * HEADS + 255) / 256, 256, 0, stream>>>(src, dst, el, er, m, E);
    gat_edge_sum<<<(E * HEADS + 255) / 256, 256, 0, stream>>>(src, dst, el, er, m, denom, E);
    gat_message<<<(E + 7) / 8, 256, 0, stream>>>(hbuf, src, dst, el, er, m, denom, out, E);
}

extern "C" void kernel_launch(void* const* d_in, const int* in_sizes, int n_in,
                              void* d_out, int out_size, void* d_ws, size_t ws_size,
                              hipStream_t stream) {
    const float* feat = (const float*)d_in[0];
    const int*   esrc = (const int*)d_in[1];
    const int*   edst = (const int*)d_in[2];
    const float* W0   = (const float*)d_in[3];
    const float* al0  = (const float*)d_in[4];
    const float* ar0  = (const float*)d_in[5];
    const float* W1   = (const float*)d_in[6];
    const float* al1  = (const float*)d_in[7];
    const float* ar1  = (const float*)d_in[8];

    int N = in_sizes[0] / IN_FEATS;
    int E = in_sizes[1];

    // workspace layout
    float*    hbuf  = (float*)d_ws;                        // N*128
    float*    x1    = hbuf + (long)N * DIM;                // N*128 (layer-1 out)
    float*    el    = x1 + (long)N * DIM;                  // N*8
    float*    er    = el + (long)N * HEADS;                // N*8
    unsigned* m     = (unsigned*)(er + (long)N * HEADS);   // N*8
    float*    denom = (float*)(m + (long)N * HEADS);       // N*8

    // layer 1: feat -> x1
    run_layer(feat, W0, al0, ar0, esrc, edst, hbuf, el, er, m, denom, x1,
              N, E, stream);
    // layer 2: x1 -> d_out
    run_layer(x1, W1, al1, ar1, esrc, edst, hbuf, el, er, m, denom,
              (float*)d_out, N, E, stream);
}